// Cons_Loss_79027398246842
// MI455X (gfx1250) — compile-verified
//
#include <hip/hip_runtime.h>

typedef __attribute__((ext_vector_type(2))) float v2f;
typedef __attribute__((ext_vector_type(8))) float v8f;

#define HH    512
#define WW    512
#define HW    (HH * WW)          // 262144
#define LL    4
#define NQ    64
#define NCOL  9                  // 4*s1, 4*s2, cnt
#define QC    (NQ * NCOL)        // 576
#define EPSF  1e-5f
#define NBLK  128
#define WPB   4                  // wave32s per block (128 threads)
#define NWAVE (NBLK * WPB)       // 512 waves == 512 image rows
#define KST   (WW / 4)           // 128 K-steps of 4 pixels per row

// ---------------------------------------------------------------------------
// Kernel 1: fused mask + WMMA reduction.  One wave per image row, so the
// row coordinate py is wave-constant: edge area = (C + py*Dy) + px*Dx is
// hoisted to a single FMA per edge per pixel.  Each wave accumulates four
// 16x16 fp32 tiles D[m,n] += mask[m,k] * val[n,k] over its row using
// V_WMMA_F32_16X16X4_F32 (full fp32 precision, K = 4 pixels / step).
//   m = quad (4 tiles of 16 -> 64 quads)
//   n = 0..3 : pred[l], 4..7 : pred[l]^2, 8 : 1.0 (cnt), 9..15 : 0
// ---------------------------------------------------------------------------
__global__ __launch_bounds__(128) void cons_loss_partial(
    const float* __restrict__ pred, const float* __restrict__ gt,
    const float* __restrict__ boxes, float* __restrict__ partials)
{
    __shared__ float sC[NQ * 4], sDx[NQ * 4], sDy[NQ * 4];
    __shared__ float sacc[WPB * QC];

    const int tid = threadIdx.x;

    // Edge coefficients: area_e(px,py) = C + py*Dy + px*Dx  (0.5 folded in)
    for (int i = tid; i < NQ * 4; i += 128) {
        int q = i >> 2, e = i & 3, e2 = (e + 1) & 3;
        float a0 = boxes[q * 8 + e * 2 + 0], a1 = boxes[q * 8 + e * 2 + 1];
        float b0 = boxes[q * 8 + e2 * 2 + 0], b1 = boxes[q * 8 + e2 * 2 + 1];
        sC[i]  = 0.5f * (a0 * b1 - b0 * a1);
        sDy[i] = 0.5f * (b0 - a0);
        sDx[i] = 0.5f * (a1 - b1);
    }
    __syncthreads();

    const int lane  = tid & 31;
    const int wv    = tid >> 5;
    const int m     = lane & 15;   // A row (quad-in-tile); also B column index
    const int khalf = lane >> 4;   // lanes 0-15: K=0,1   lanes 16-31: K=2,3
    const int n     = m;

    const int row    = blockIdx.x * WPB + wv;     // this wave's image row
    const float py   = (float)row;
    const int pstart = row * WW;

    // Loop-invariant, py-folded per-lane coefficients -> 32 registers
    float c2[4][4], dxc[4][4];
#pragma unroll
    for (int t = 0; t < 4; ++t)
#pragma unroll
        for (int e = 0; e < 4; ++e) {
            int idx = (t * 16 + m) * 4 + e;
            c2[t][e]  = fmaf(py, sDy[idx], sC[idx]);   // C + py*Dy
            dxc[t][e] = sDx[idx];
        }

    // mask = all(area>=0) && |area_even|>=eps && gt>0
    //      = a0>=eps && a1>=0 && a2>=eps && a3>=0 && gt>0
    auto maskval = [&](int t, float px, float g) -> float {
        float ar0 = fmaf(px, dxc[t][0], c2[t][0]);
        float ar1 = fmaf(px, dxc[t][1], c2[t][1]);
        float ar2 = fmaf(px, dxc[t][2], c2[t][2]);
        float ar3 = fmaf(px, dxc[t][3], c2[t][3]);
        bool ok = (ar0 >= EPSF) & (ar1 >= 0.0f) & (ar2 >= EPSF) &
                  (ar3 >= 0.0f) & (g > 0.0f);
        return ok ? 1.0f : 0.0f;
    };

    v8f zero8 = {0.f, 0.f, 0.f, 0.f, 0.f, 0.f, 0.f, 0.f};
    v8f acc[4];
#pragma unroll
    for (int t = 0; t < 4; ++t) acc[t] = zero8;

    const int layer = n & 3;
    const float* predl = pred + layer * HW;

    for (int s = 0; s < KST; ++s) {
        int xA = s * 4 + khalf * 2;     // this lane's two K-pixels (columns)
        int xB = xA + 1;
        int pA = pstart + xA;
        int pB = pA + 1;
        float pxA = (float)xA, pxB = (float)xB;
        float gA = gt[pA], gB = gt[pB];

        __builtin_prefetch(predl + pA + 256, 0, 3);  // global_prefetch_b8

        // B matrix (4x16 fp32): v2f = {K=2*khalf, K=2*khalf+1} rows, col n
        float v0 = predl[pA], v1 = predl[pB];
        float bA = (n < 4) ? v0 : ((n < 8) ? v0 * v0 : ((n == 8) ? 1.0f : 0.0f));
        float bB = (n < 4) ? v1 : ((n < 8) ? v1 * v1 : ((n == 8) ? 1.0f : 0.0f));
        v2f bvec; bvec[0] = bA; bvec[1] = bB;

        // A matrices (16x4 fp32 mask per tile) + fp32 WMMA accumulate
#pragma unroll
        for (int t = 0; t < 4; ++t) {
            v2f av;
            av[0] = maskval(t, pxA, gA);
            av[1] = maskval(t, pxB, gB);
            acc[t] = __builtin_amdgcn_wmma_f32_16x16x4_f32(
                false, av, false, bvec, (short)0, acc[t], false, false);
        }
    }

    // D layout: VGPR j -> M = khalf*8 + j, N = lane&15.  Stage to LDS,
    // reduce the 4 waves, emit one deterministic [64][9] partial per block.
#pragma unroll
    for (int t = 0; t < 4; ++t)
#pragma unroll
        for (int j = 0; j < 8; ++j) {
            int mrow = t * 16 + khalf * 8 + j;
            if (n < NCOL) sacc[wv * QC + mrow * NCOL + n] = acc[t][j];
        }
    __syncthreads();

    for (int i = tid; i < QC; i += 128) {
        float s = sacc[i] + sacc[QC + i] + sacc[2 * QC + i] + sacc[3 * QC + i];
        partials[blockIdx.x * QC + i] = s;
    }
}

// ---------------------------------------------------------------------------
// Kernel 2: reduce 128 block-partials, apply variance formula, scalar out.
// ---------------------------------------------------------------------------
__global__ __launch_bounds__(256) void cons_loss_final(
    const float* __restrict__ partials, float* __restrict__ out)
{
    __shared__ float red[QC];
    __shared__ float qloss[NQ];
    const int tid = threadIdx.x;

    for (int i = tid; i < QC; i += 256) {
        float s = 0.0f;
        for (int b = 0; b < NBLK; ++b) s += partials[b * QC + i];
        red[i] = s;
    }
    __syncthreads();

    if (tid < NQ) {
        int q = tid;
        float cnt  = red[q * NCOL + 8];
        float safe = fmaxf(cnt, 1.0f);
        float a = 0.0f;
        for (int l = 0; l < LL; ++l) {
            float s1 = red[q * NCOL + l];
            float s2 = red[q * NCOL + 4 + l];
            float mean = s1 / safe;
            a += (s2 - 2.0f * mean * s1 + mean * mean * cnt) / safe;
        }
        qloss[q] = (cnt > 0.0f) ? a : 0.0f;
    }
    __syncthreads();

    if (tid == 0) {
        float s = 0.0f;
        for (int q = 0; q < NQ; ++q) s += qloss[q];
        out[0] = s;
    }
}

extern "C" void kernel_launch(void* const* d_in, const int* in_sizes, int n_in,
                              void* d_out, int out_size, void* d_ws, size_t ws_size,
                              hipStream_t stream) {
    (void)in_sizes; (void)n_in; (void)out_size; (void)ws_size;
    const float* pred  = (const float*)d_in[0];   // [1,4,512,512]
    const float* gt    = (const float*)d_in[1];   // [1,512,512]
    const float* boxes = (const float*)d_in[2];   // [64,4,2]
    float* partials = (float*)d_ws;               // 128 * 576 floats = 295 KB
    float* out = (float*)d_out;

    cons_loss_partial<<<NBLK, 128, 0, stream>>>(pred, gt, boxes, partials);
    cons_loss_final<<<1, 256, 0, stream>>>(partials, out);
}